// K_WTA_81381040324887
// MI455X (gfx1250) — compile-verified
//
#include <hip/hip_runtime.h>
#include <stdint.h>

// ---------------- problem constants ----------------
#define NROWS        64
#define ROW_ELEMS    (256 * 2048)        // 524288 floats per row
#define ROW_CHUNKS   (ROW_ELEMS / 4)     // 131072 float4 per row
#define KSEL         52428               // k-th largest (gamma=0.1)
#define NBINS        2048
#define SEL_THREADS  1024                // 32 waves (wave32)
#define BPR          16                  // blocks per row in histogram pass
#define SEG_CHUNKS   (ROW_CHUNKS / BPR)  // 8192 float4 per block
#define MASK_THREADS 256                 // 8 waves
#define TOTAL_CHUNKS (NROWS * ROW_CHUNKS)          // 8388608
#define TOTAL_TILES  (TOTAL_CHUNKS / MASK_THREADS) // 32768 tiles of 4KB
#define MASK_BLOCKS  2048
#define ROW_CHUNK_SHIFT 17               // float4-chunks per row = 1<<17

typedef float v4f __attribute__((ext_vector_type(4)));
typedef int   v4i __attribute__((vector_size(16)));   // matches builtin param type

// ---------------- monotonic float<->key ----------------
__device__ __forceinline__ unsigned key_of(float f) {
  unsigned u = __float_as_uint(f);
  return (u & 0x80000000u) ? ~u : (u | 0x80000000u);
}
__device__ __forceinline__ float float_of_key(unsigned k) {
  unsigned u = (k & 0x80000000u) ? (k ^ 0x80000000u) : ~k;
  return __uint_as_float(u);
}

// ---------------- CDNA5 async-to-LDS helpers ----------------
__device__ __forceinline__ void async_ld_b128(void* lds_dst, const void* gsrc) {
#if __has_builtin(__builtin_amdgcn_global_load_async_to_lds_b128)
  __builtin_amdgcn_global_load_async_to_lds_b128(
      (__attribute__((address_space(1))) v4i*)gsrc,
      (__attribute__((address_space(3))) v4i*)lds_dst,
      0, 0);
#else
  // GV mode: vdst = LDS byte offset, vaddr = 64-bit global address
  unsigned loff = (unsigned)(uintptr_t)lds_dst;
  asm volatile("global_load_async_to_lds_b128 %0, %1, off"
               :: "v"(loff), "v"(gsrc) : "memory");
#endif
}

template <int N>
__device__ __forceinline__ void wait_async() {
#if __has_builtin(__builtin_amdgcn_s_wait_asynccnt)
  __builtin_amdgcn_s_wait_asynccnt(N);
#else
  asm volatile("s_wait_asynccnt %0" :: "i"(N) : "memory");
#endif
}

// ---------------- init: zero global hist, init per-row state ----------------
__global__ void wta_init(unsigned* __restrict__ ghist, unsigned* __restrict__ state) {
  const size_t i = (size_t)blockIdx.x * blockDim.x + threadIdx.x;
  const size_t total = (size_t)NROWS * NBINS;
  if (i < total) ghist[i] = 0u;
  if (i < NROWS) { state[2 * i] = 0u; state[2 * i + 1] = (unsigned)KSEL; }
}

// ---------------- histogram accumulate (one radix pass) ----------------
__global__ __launch_bounds__(SEL_THREADS)
void wta_hist(const float* __restrict__ x, unsigned* __restrict__ ghist,
              const unsigned* __restrict__ state, int pass) {
  __shared__ unsigned hist[NBINS];
  const int row = blockIdx.x / BPR;
  const int seg = blockIdx.x % BPR;
  const int tid = threadIdx.x;

  for (int b = tid; b < NBINS; b += SEL_THREADS) hist[b] = 0u;
  __syncthreads();

  int lo, width;
  if (pass == 0)      { lo = 21; width = 11; }
  else if (pass == 1) { lo = 10; width = 11; }
  else                { lo = 0;  width = 10; }
  const int ms = lo + width;                 // bits already determined above field
  const unsigned fmask = (1u << width) - 1u;
  const unsigned pfx = state[2 * row];
  const unsigned pfxHi = (pass == 0) ? 0u : (pfx >> ms);

  const v4f* __restrict__ xr =
      (const v4f*)(x + (size_t)row * ROW_ELEMS) + (size_t)seg * SEG_CHUNKS;

  for (int i = tid; i < SEG_CHUNKS; i += SEL_THREADS) {
    __builtin_prefetch(xr + i + 4 * SEL_THREADS, 0, 1);  // global_prefetch_b8
    v4f v = xr[i];
    unsigned k0 = key_of(v.x), k1 = key_of(v.y), k2 = key_of(v.z), k3 = key_of(v.w);
    if (pass == 0 || (k0 >> ms) == pfxHi) atomicAdd(&hist[(k0 >> lo) & fmask], 1u);
    if (pass == 0 || (k1 >> ms) == pfxHi) atomicAdd(&hist[(k1 >> lo) & fmask], 1u);
    if (pass == 0 || (k2 >> ms) == pfxHi) atomicAdd(&hist[(k2 >> lo) & fmask], 1u);
    if (pass == 0 || (k3 >> ms) == pfxHi) atomicAdd(&hist[(k3 >> lo) & fmask], 1u);
  }
  __syncthreads();

  unsigned* __restrict__ gh = ghist + (size_t)row * NBINS;
  for (int b = tid; b < NBINS; b += SEL_THREADS) {
    unsigned c = hist[b];
    if (c) atomicAdd(&gh[b], c);
  }
}

// ---------------- pick bin: suffix scan + refine prefix ----------------
__global__ __launch_bounds__(SEL_THREADS)
void wta_pick(unsigned* __restrict__ ghist, unsigned* __restrict__ state,
              float* __restrict__ thr, int pass) {
  __shared__ unsigned S[NBINS];
  __shared__ unsigned s_bin, s_above;
  const int row = blockIdx.x;
  const int tid = threadIdx.x;
  const int b0 = tid, b1 = tid + SEL_THREADS;

  unsigned* __restrict__ gh = ghist + (size_t)row * NBINS;
  S[b0] = gh[b0];  S[b1] = gh[b1];
  gh[b0] = 0u;     gh[b1] = 0u;          // ready for next pass
  __syncthreads();

  // in-place inclusive suffix-sum from the top: S[b] = sum_{j>=b} hist[j]
  for (int d = 1; d < NBINS; d <<= 1) {
    unsigned a0 = (b0 + d < NBINS) ? S[b0 + d] : 0u;
    unsigned a1 = (b1 + d < NBINS) ? S[b1 + d] : 0u;
    __syncthreads();
    S[b0] += a0;  S[b1] += a1;
    __syncthreads();
  }

  const unsigned rk = state[2 * row + 1];
  {
    unsigned S0 = S[b0], N0 = S[b0 + 1];                       // b0+1 <= 1024
    unsigned S1 = S[b1], N1 = (b1 + 1 < NBINS) ? S[b1 + 1] : 0u;
    if (S0 >= rk && N0 < rk) { s_bin = (unsigned)b0; s_above = N0; }
    if (S1 >= rk && N1 < rk) { s_bin = (unsigned)b1; s_above = N1; }
  }
  __syncthreads();

  if (tid == 0) {
    const int lo = (pass == 0) ? 21 : ((pass == 1) ? 10 : 0);
    const unsigned pfx = state[2 * row] | (s_bin << lo);
    state[2 * row]     = pfx;
    state[2 * row + 1] = rk - s_above;
    if (pass == 2) thr[row] = float_of_key(pfx);  // exact k-th largest value
  }
}

// ---------------- mask: async double-buffered stream through LDS ----------------
__global__ __launch_bounds__(MASK_THREADS)
void wta_mask(const float* __restrict__ x, const float* __restrict__ thr,
              float* __restrict__ out) {
  __shared__ v4f buf[2][MASK_THREADS];   // 8 KB, each wave owns its 32-lane slice
  const int tid = threadIdx.x;
  const v4f* __restrict__ x4 = (const v4f*)x;
  v4f* __restrict__ o4 = (v4f*)out;
  const int stride = (int)gridDim.x;

  int t = (int)blockIdx.x;
  int stage = 0;
  if (t < TOTAL_TILES)
    async_ld_b128(&buf[0][tid], &x4[(size_t)t * MASK_THREADS + tid]);

  for (; t < TOTAL_TILES; t += stride) {
    const int nxt = t + stride;
    if (nxt < TOTAL_TILES) {
      async_ld_b128(&buf[stage ^ 1][tid], &x4[(size_t)nxt * MASK_THREADS + tid]);
      wait_async<1>();   // oldest (tile t) complete; loads retire in order
    } else {
      wait_async<0>();
    }
    asm volatile("" ::: "memory");

    const size_t g = (size_t)t * MASK_THREADS + tid;
    v4f v = buf[stage][tid];
    const float T = thr[g >> ROW_CHUNK_SHIFT];   // tile never straddles a row
    v4f o;
    o.x = (v.x < T) ? v.x : 0.0f;
    o.y = (v.y < T) ? v.y : 0.0f;
    o.z = (v.z < T) ? v.z : 0.0f;
    o.w = (v.w < T) ? v.w : 0.0f;
    __builtin_nontemporal_store(o, &o4[g]);      // output never re-read: NT store
    stage ^= 1;
  }
}

// ---------------- launch ----------------
extern "C" void kernel_launch(void* const* d_in, const int* in_sizes, int n_in,
                              void* d_out, int out_size, void* d_ws, size_t ws_size,
                              hipStream_t stream) {
  (void)in_sizes; (void)n_in; (void)out_size; (void)ws_size;
  const float* x = (const float*)d_in[0];
  float* out = (float*)d_out;

  // Scratch: 512 KB of per-row histograms lives at the front of d_out (the mask
  // pass overwrites all of d_out afterwards). d_ws holds select state + thresholds.
  unsigned* ghist = (unsigned*)d_out;
  unsigned* state = (unsigned*)d_ws;                    // 2 u32 per row
  float* thr = (float*)((char*)d_ws + 1024);            // 64 floats

  wta_init<<<(NROWS * NBINS + 255) / 256, 256, 0, stream>>>(ghist, state);
  for (int pass = 0; pass < 3; ++pass) {
    wta_hist<<<NROWS * BPR, SEL_THREADS, 0, stream>>>(x, ghist, state, pass);
    wta_pick<<<NROWS, SEL_THREADS, 0, stream>>>(ghist, state, thr, pass);
  }
  wta_mask<<<MASK_BLOCKS, MASK_THREADS, 0, stream>>>(x, thr, out);
}